// MACE_54881092108427
// MI455X (gfx1250) — compile-verified
//
#include <hip/hip_runtime.h>
#include <math.h>

#define Nn   32768
#define Ee   262144
#define Ff   64
#define Zz   10
#define NBb  8
#define Ll   2
#define Hh   16
#define NF   (Nn * Ff)

typedef __attribute__((ext_vector_type(2))) float v2f;
typedef __attribute__((ext_vector_type(8))) float v8f;
typedef unsigned int u32;
typedef __attribute__((ext_vector_type(4))) u32 tdm_v4u;
typedef __attribute__((ext_vector_type(8))) int tdm_v8i;
typedef __attribute__((ext_vector_type(4))) int tdm_v4i;

#if defined(__has_builtin)
#if __has_builtin(__builtin_amdgcn_tensor_load_to_lds) && \
    __has_builtin(__builtin_amdgcn_s_wait_tensorcnt)
#define USE_TDM 1
#endif
#endif

// ---------------------------------------------------------------------------
// Edge geometry: r, Bessel radial basis with polynomial envelope, Y1 = v/r
// ---------------------------------------------------------------------------
__global__ void geom_kernel(const float* __restrict__ vec,
                            float* __restrict__ rbf,
                            float* __restrict__ y1) {
  int e = blockIdx.x * blockDim.x + threadIdx.x;
  if (e >= Ee) return;
  float x = vec[3 * e + 0], y = vec[3 * e + 1], z = vec[3 * e + 2];
  float r = sqrtf(x * x + y * y + z * z + 1e-12f);
  float inv = 1.0f / r;
  y1[3 * e + 0] = x * inv;
  y1[3 * e + 1] = y * inv;
  y1[3 * e + 2] = z * inv;
  float rc = fmaxf(r, 1e-6f);
  float env = 0.0f;
  if (r < 1.0f) {
    // p=6: 1 - 28 r^6 + 48 r^7 - 21 r^8
    float r2 = r * r, r3 = r2 * r, r6 = r3 * r3, r7 = r6 * r, r8 = r7 * r;
    env = 1.0f - 28.0f * r6 + 48.0f * r7 - 21.0f * r8;
  }
  float sc = 1.41421356237309f * env / rc;
#pragma unroll
  for (int n = 1; n <= NBb; ++n)
    rbf[e * NBb + (n - 1)] = sc * __sinf((float)n * 3.14159265358979f * rc);
}

// ---------------------------------------------------------------------------
// Node embedding: s[n,f] = embed_s[species[n], f]
// ---------------------------------------------------------------------------
__global__ void embed_kernel(const float* __restrict__ embed_s,
                             const int* __restrict__ species,
                             float* __restrict__ s) {
  int idx = blockIdx.x * blockDim.x + threadIdx.x;
  if (idx >= NF) return;
  int n = idx >> 6;
  int c = idx & 63;
  s[idx] = embed_s[species[n] * Ff + c];
}

// ---------------------------------------------------------------------------
// Fused edge kernel: radial MLP + gather + CG tensor product + atomic scatter
// 256 threads = 4 edges x 64 channels. v stored planar: [3][N][F].
// ---------------------------------------------------------------------------
__global__ void edge_msg_kernel(const float* __restrict__ rbf,
                                const float* __restrict__ y1,
                                const float* __restrict__ s_cur,
                                const float* __restrict__ v_cur,
                                const float* __restrict__ WrI,
                                const int* __restrict__ senders,
                                const int* __restrict__ receivers,
                                float* __restrict__ agg_s,
                                float* __restrict__ agg_v) {
  __shared__ float sh_rbf[4][NBb];
  __shared__ float sh_y[4][3];
  int el = threadIdx.x >> 6;
  int f  = threadIdx.x & 63;
  int e  = blockIdx.x * 4 + el;
  if (f < NBb)
    sh_rbf[el][f] = rbf[e * NBb + f];
  else if (f < NBb + 3)
    sh_y[el][f - NBb] = y1[e * 3 + (f - NBb)];
  __syncthreads();

  // radial MLP: 5 path weights per channel, rbf[8] @ Wr[8,5*64]
  float w0 = 0.f, w1 = 0.f, w2 = 0.f, w3 = 0.f, w4 = 0.f;
#pragma unroll
  for (int n = 0; n < NBb; ++n) {
    float rb = sh_rbf[el][n];
    const float* Wn = WrI + n * 5 * Ff + f;
    w0 += rb * Wn[0 * Ff];
    w1 += rb * Wn[1 * Ff];
    w2 += rb * Wn[2 * Ff];
    w3 += rb * Wn[3 * Ff];
    w4 += rb * Wn[4 * Ff];
  }
  float yx = sh_y[el][0], yy = sh_y[el][1], yz = sh_y[el][2];
  int snd = senders[e];
  int rcv = receivers[e];
  float ss = s_cur[snd * Ff + f];
  float vx = v_cur[0 * NF + snd * Ff + f];
  float vy = v_cur[1 * NF + snd * Ff + f];
  float vz = v_cur[2 * NF + snd * Ff + f];
  float dot = vx * yx + vy * yy + vz * yz;
  const float EPSs = 0.24253562503633297f;  // 1/sqrt(17)
  float m0 = (w0 * ss + w1 * dot) * EPSs;
  float cx = vy * yz - vz * yy;
  float cy = vz * yx - vx * yz;
  float cz = vx * yy - vy * yx;
  float b2 = w2 * ss;
  float mx = (b2 * yx + w3 * vx + w4 * cx) * EPSs;
  float my = (b2 * yy + w3 * vy + w4 * cy) * EPSs;
  float mz = (b2 * yz + w3 * vz + w4 * cz) * EPSs;
  unsafeAtomicAdd(&agg_s[rcv * Ff + f], m0);
  unsafeAtomicAdd(&agg_v[0 * NF + rcv * Ff + f], mx);
  unsafeAtomicAdd(&agg_v[1 * NF + rcv * Ff + f], my);
  unsafeAtomicAdd(&agg_v[2 * NF + rcv * Ff + f], mz);
}

// ---------------------------------------------------------------------------
// D[N,64] = A[N,64] @ W[64,64] (+ Cin), batched over blockIdx.y with
// planeStride applied to A/Cin/D (same W each plane).
// W staged in LDS via Tensor Data Mover (1D tensor, 4096 f32, whole-tile).
// Block = 128 threads = 4 waves; each wave -> 16 rows x 64 cols
// via 4 col-tiles x 16 K-steps of v_wmma_f32_16x16x4_f32.
// A frag (16x4 f32): lane m = lane&15, K-pair base = (lane>>4)*2.
// C/D (16x16 f32): row = vgpr + 8*(lane>>4), col = lane&15.
// ---------------------------------------------------------------------------
__global__ void gemm64_kernel(const float* __restrict__ A,
                              const float* __restrict__ W,
                              const float* __restrict__ Cin,
                              float* __restrict__ D,
                              unsigned int planeStride) {
  __shared__ float Wlds[Ff * Ff];
  int tid = threadIdx.x;
  size_t poff = (size_t)blockIdx.y * planeStride;
  A += poff;
  D += poff;

#ifdef USE_TDM
  // --- TDM: async DMA of the 16KB weight matrix into LDS (wave 0 issues) ---
  if (tid < 32) {
    unsigned long long ga = (unsigned long long)(uintptr_t)W;
    u32 lds_off = (u32)(uintptr_t)&Wlds[0];  // LDS aperture: low 32 bits
    // D# group 0: count=1 | lds_addr | global_addr[56:0] | type=2
    tdm_v4u g0 = {1u, lds_off, (u32)(ga & 0xFFFFFFFFu),
                  (u32)((ga >> 32) & 0x01FFFFFFu) | 0x80000000u};
    // D# group 1: data_size=4B; tensor_dim0=4096, tensor_dim1=1;
    //             tile_dim0=4096, tile_dim1=1; tensor_dim0_stride=4096
    tdm_v8i g1 = {0x00020000, 0x10000000, 0x00010000, 0x10000000,
                  1, 4096, 0, 0};
    tdm_v4i gz4 = {0, 0, 0, 0};
    tdm_v8i gz8 = {0, 0, 0, 0, 0, 0, 0, 0};
    __builtin_amdgcn_tensor_load_to_lds(g0, g1, gz4, gz4, gz8, 0);
    __builtin_amdgcn_s_wait_tensorcnt(0);
  }
  __syncthreads();
#else
  // fallback: cooperative 16KB weight load (float4)
  {
    const float4* Wv = (const float4*)W;
    float4* Wl = (float4*)Wlds;
    for (int i = tid; i < (Ff * Ff) / 4; i += 128) Wl[i] = Wv[i];
    __syncthreads();
  }
#endif

  int wave = tid >> 5;
  int lane = tid & 31;
  int row0 = blockIdx.x * 64 + wave * 16;
  int m    = lane & 15;
  int hi   = lane >> 4;
  int kb   = hi * 2;

  // preload all 16 A fragments for this 16-row strip
  v2f afrag[16];
  const float* Arow = A + (size_t)(row0 + m) * Ff;
#pragma unroll
  for (int t = 0; t < 16; ++t) {
    afrag[t].x = Arow[t * 4 + kb + 0];
    afrag[t].y = Arow[t * 4 + kb + 1];
  }

#pragma unroll
  for (int j = 0; j < 4; ++j) {
    int col = j * 16 + m;
    v8f acc = {};
    if (Cin) {
#pragma unroll
      for (int r = 0; r < 8; ++r)
        acc[r] = Cin[poff + (size_t)(row0 + r + hi * 8) * Ff + col];
    }
#pragma unroll
    for (int t = 0; t < 16; ++t) {
      v2f b;
      b.x = Wlds[(t * 4 + kb + 0) * Ff + col];
      b.y = Wlds[(t * 4 + kb + 1) * Ff + col];
      acc = __builtin_amdgcn_wmma_f32_16x16x4_f32(
          /*neg_a=*/false, afrag[t], /*neg_b=*/false, b,
          /*c_mod=*/(short)0, acc, /*reuse_a=*/false, /*reuse_b=*/false);
    }
#pragma unroll
    for (int r = 0; r < 8; ++r)
      D[(size_t)(row0 + r + hi * 8) * Ff + col] = acc[r];
  }
}

// ---------------------------------------------------------------------------
// Node elementwise: per-species skip connection (VALU dots) + symmetric
// product basis up to correlation 3. 256 threads = 4 nodes x 64 channels.
// ---------------------------------------------------------------------------
__global__ void node_update_kernel(const float* __restrict__ s2,
                                   const float* __restrict__ v2,
                                   const float* __restrict__ s_cur,
                                   const float* __restrict__ v_cur,
                                   const float* __restrict__ skip_s,
                                   const float* __restrict__ skip_v,
                                   const float* __restrict__ pwI,
                                   const int* __restrict__ species,
                                   float* __restrict__ ps,
                                   float* __restrict__ pv,
                                   float* __restrict__ sc_s,
                                   float* __restrict__ sc_v,
                                   int doSkip) {
  __shared__ float sh_s[4][Ff];
  __shared__ float sh_v[3][4][Ff];
  int nl = threadIdx.x >> 6;
  int c  = threadIdx.x & 63;
  int n  = blockIdx.x * 4 + nl;
  int sp = species[n];

  if (doSkip) {
    sh_s[nl][c]    = s_cur[n * Ff + c];
    sh_v[0][nl][c] = v_cur[0 * NF + n * Ff + c];
    sh_v[1][nl][c] = v_cur[1 * NF + n * Ff + c];
    sh_v[2][nl][c] = v_cur[2 * NF + n * Ff + c];
    __syncthreads();
    const float* SS = skip_s + (size_t)sp * Ff * Ff;
    const float* SV = skip_v + (size_t)sp * Ff * Ff;
    float as = 0.f, a0 = 0.f, a1 = 0.f, a2 = 0.f;
#pragma unroll 8
    for (int fch = 0; fch < Ff; ++fch) {
      float wS = SS[fch * Ff + c];
      float wV = SV[fch * Ff + c];
      as += sh_s[nl][fch] * wS;
      a0 += sh_v[0][nl][fch] * wV;
      a1 += sh_v[1][nl][fch] * wV;
      a2 += sh_v[2][nl][fch] * wV;
    }
    sc_s[n * Ff + c] = as;
    sc_v[0 * NF + n * Ff + c] = a0;
    sc_v[1 * NF + n * Ff + c] = a1;
    sc_v[2 * NF + n * Ff + c] = a2;
  }

  // symmetric product basis
  float x  = s2[n * Ff + c];
  float vx = v2[0 * NF + n * Ff + c];
  float vy = v2[1 * NF + n * Ff + c];
  float vz = v2[2 * NF + n * Ff + c];
  float vv = vx * vx + vy * vy + vz * vz;
  const float* p = pwI + (size_t)sp * 9 * Ff + c;
  float p0 = p[0 * Ff], p1 = p[1 * Ff], p2 = p[2 * Ff], p3 = p[3 * Ff];
  float p4 = p[4 * Ff], p5 = p[5 * Ff], p6 = p[6 * Ff], p7 = p[7 * Ff];
  float p8 = p[8 * Ff];
  float x2 = x * x;
  float pss = p0 * x + p1 * x2 + p2 * vv + p3 * x2 * x + p4 * x * vv;
  float pvf = p5 + p6 * x + p7 * x2 + p8 * vv;
  ps[n * Ff + c] = pss;
  pv[0 * NF + n * Ff + c] = pvf * vx;
  pv[1 * NF + n * Ff + c] = pvf * vy;
  pv[2 * NF + n * Ff + c] = pvf * vz;
}

// ---------------------------------------------------------------------------
// Readouts: out[n,0] = s0@Wread0 ; out[n,1] = silu(s1@Wr1a)@Wr1b
// ---------------------------------------------------------------------------
__global__ void readout_kernel(const float* __restrict__ s0,
                               const float* __restrict__ s1,
                               const float* __restrict__ W0,
                               const float* __restrict__ Wa,
                               const float* __restrict__ Wb,
                               float* __restrict__ out) {
  __shared__ float shW0[Ff];
  __shared__ float shA[Ff * Hh];
  __shared__ float shB[Hh];
  int tid = threadIdx.x;
  if (tid < Ff) shW0[tid] = W0[tid];
  if (tid < Hh) shB[tid] = Wb[tid];
  for (int i = tid; i < Ff * Hh; i += 256) shA[i] = Wa[i];
  __syncthreads();

  int n = blockIdx.x * 256 + tid;
  const float* r0 = s0 + (size_t)n * Ff;
  float o0 = 0.f;
#pragma unroll 8
  for (int fch = 0; fch < Ff; ++fch) o0 += r0[fch] * shW0[fch];
  out[n * 2 + 0] = o0;

  float h[Hh];
#pragma unroll
  for (int j = 0; j < Hh; ++j) h[j] = 0.f;
  const float* r1 = s1 + (size_t)n * Ff;
  for (int fch = 0; fch < Ff; ++fch) {
    float sv = r1[fch];
#pragma unroll
    for (int j = 0; j < Hh; ++j) h[j] += sv * shA[fch * Hh + j];
  }
  float o1 = 0.f;
#pragma unroll
  for (int j = 0; j < Hh; ++j) {
    float zz = h[j];
    float sg = 1.f / (1.f + __expf(-zz));
    o1 += zz * sg * shB[j];
  }
  out[n * 2 + 1] = o1;
}

// ---------------------------------------------------------------------------
extern "C" void kernel_launch(void* const* d_in, const int* in_sizes, int n_in,
                              void* d_out, int out_size, void* d_ws, size_t ws_size,
                              hipStream_t stream) {
  const float* vectors = (const float*)d_in[0];
  const float* embed_s = (const float*)d_in[1];
  const float* Wr      = (const float*)d_in[2];
  const float* Wls     = (const float*)d_in[3];
  const float* Wlv     = (const float*)d_in[4];
  const float* skip_s  = (const float*)d_in[5];
  const float* skip_v  = (const float*)d_in[6];
  const float* pw      = (const float*)d_in[7];
  const float* Wps     = (const float*)d_in[8];
  const float* Wpv     = (const float*)d_in[9];
  const float* Wread0  = (const float*)d_in[10];
  const float* Wr1a    = (const float*)d_in[11];
  const float* Wr1b    = (const float*)d_in[12];
  const int* senders   = (const int*)d_in[13];
  const int* receivers = (const int*)d_in[14];
  const int* species   = (const int*)d_in[15];
  float* out = (float*)d_out;

  float* ws = (float*)d_ws;
  size_t off = 0;
  float* rbf  = ws + off; off += (size_t)Ee * NBb;
  float* y1   = ws + off; off += (size_t)Ee * 3;
  float* sA   = ws + off; off += (size_t)NF;
  float* sB   = ws + off; off += (size_t)NF;
  float* vA   = ws + off; off += (size_t)3 * NF;
  float* vB   = ws + off; off += (size_t)3 * NF;
  float* aggs = ws + off; off += (size_t)NF;
  float* aggv = ws + off; off += (size_t)3 * NF;
  float* s2b  = ws + off; off += (size_t)NF;
  float* v2b  = ws + off; off += (size_t)3 * NF;
  float* psb  = ws + off; off += (size_t)NF;
  float* pvb  = ws + off; off += (size_t)3 * NF;
  float* scs  = ws + off; off += (size_t)NF;
  float* scv  = ws + off; off += (size_t)3 * NF;

  geom_kernel<<<Ee / 256, 256, 0, stream>>>(vectors, rbf, y1);
  embed_kernel<<<NF / 256, 256, 0, stream>>>(embed_s, species, sA);
  (void)hipMemsetAsync(vA, 0, (size_t)3 * NF * sizeof(float), stream);

  dim3 g1(Nn / 64, 1), g3(Nn / 64, 3);

  for (int i = 0; i < Ll; ++i) {
    const float* s_cur = (i == 0) ? sA : sB;
    const float* v_cur = (i == 0) ? vA : vB;
    float* s_nxt = (i == 0) ? sB : sA;
    float* v_nxt = (i == 0) ? vB : vA;

    (void)hipMemsetAsync(aggs, 0, (size_t)NF * sizeof(float), stream);
    (void)hipMemsetAsync(aggv, 0, (size_t)3 * NF * sizeof(float), stream);

    edge_msg_kernel<<<Ee / 4, 256, 0, stream>>>(
        rbf, y1, s_cur, v_cur, Wr + (size_t)i * NBb * 5 * Ff,
        senders, receivers, aggs, aggv);

    gemm64_kernel<<<g1, 128, 0, stream>>>(aggs, Wls + (size_t)i * Ff * Ff,
                                          nullptr, s2b, 0u);
    gemm64_kernel<<<g3, 128, 0, stream>>>(aggv, Wlv + (size_t)i * Ff * Ff,
                                          nullptr, v2b, (unsigned)NF);

    node_update_kernel<<<Nn / 4, 256, 0, stream>>>(
        s2b, v2b, s_cur, v_cur, skip_s, skip_v,
        pw + (size_t)i * Zz * 9 * Ff, species, psb, pvb, scs, scv,
        (i > 0) ? 1 : 0);

    gemm64_kernel<<<g1, 128, 0, stream>>>(
        psb, Wps + (size_t)i * Ff * Ff, (i > 0) ? scs : nullptr, s_nxt, 0u);
    gemm64_kernel<<<g3, 128, 0, stream>>>(
        pvb, Wpv + (size_t)i * Ff * Ff, (i > 0) ? scv : nullptr, v_nxt,
        (unsigned)NF);
  }

  // readout 0 uses s after interaction 0 (sB); readout 1 uses final s (sA)
  readout_kernel<<<Nn / 256, 256, 0, stream>>>(sB, sA, Wread0, Wr1a, Wr1b, out);
}